// GraphNet_76854144795115
// MI455X (gfx1250) — compile-verified
//
#include <hip/hip_runtime.h>
#include <hip/hip_bf16.h>

typedef float v2f __attribute__((ext_vector_type(2)));
typedef float v8f __attribute__((ext_vector_type(8)));

#define NG   128
#define NPG  1024
#define NN   (NG * NPG)
#define KNN  16
#define SLOPE 0.01f

__device__ __forceinline__ float leaky(float v) { return v >= 0.0f ? v : SLOPE * v; }

__device__ __forceinline__ v8f wmma4(v2f a, v2f b, v8f c) {
  // V_WMMA_F32_16X16X4_F32 : D(16x16 f32) = A(16x4 f32) * B(4x16 f32) + C
  return __builtin_amdgcn_wmma_f32_16x16x4_f32(
      /*neg_a=*/false, a, /*neg_b=*/false, b,
      /*c_mod=*/(short)0, c, /*reuse_a=*/false, /*reuse_b=*/false);
}

// ---------------------------------------------------------------------------
// kNN: per graph, Gram-matrix distances via WMMA + per-lane top-16 selection.
// Ordering key per row i: sq_j - 2*dot(p_i,p_j)  (sq_i is row-constant).
// Block = 256 thr (8 waves); wave handles 32 rows; grid = NG*4 blocks.
// A-form pairs [x,y,z,1] and B-form pairs [-2x,-2y,-2z,sq] are precomputed
// in LDS as per-half float2 so the inner loop is load_b64 + 2 WMMA + store.
// dt is wave-private: wave32 lockstep + in-order DS => no workgroup barrier.
// ---------------------------------------------------------------------------
__global__ __launch_bounds__(256) void knn_kernel(const float* __restrict__ pos,
                                                  int stride,
                                                  int* __restrict__ idx_out) {
  __shared__ v2f   apos[NPG][2];        // [i][half] = {comp(2h), comp(2h+1)} of [x,y,z,1]
  __shared__ v2f   bpos[NPG][2];        // [i][half] of [-2x,-2y,-2z,sq]
  __shared__ float dt[8][32 * 17];      // per-wave 32x16 tile (stride 17)

  const int g     = blockIdx.x >> 2;
  const int chunk = blockIdx.x & 3;
  const int tid   = threadIdx.x;
  const int wave  = tid >> 5;
  const int lane  = tid & 31;
  const int half  = lane >> 4;
  const int l16   = lane & 15;

  for (int i = tid; i < NPG; i += 256) {
    const float* p = pos + (size_t)(g * NPG + i) * stride;
    float x = p[0], y = p[1], z = p[2];
    float sq = x * x + y * y + z * z;
    apos[i][0] = (v2f){x, y};
    apos[i][1] = (v2f){z, 1.0f};
    bpos[i][0] = (v2f){-2.0f * x, -2.0f * y};
    bpos[i][1] = (v2f){-2.0f * z, sq};
  }
  __syncthreads();

  const int rbase = chunk * 256 + wave * 32;
  v2f a0 = apos[rbase + l16][half];
  v2f a1 = apos[rbase + 16 + l16][half];

  float bd[KNN];
  int   bi[KNN];
#pragma unroll
  for (int m = 0; m < KNN; ++m) { bd[m] = 3.0e38f; bi[m] = 0; }

  const int rloc = rbase + lane;        // row this lane selects for

  for (int ct = 0; ct < 64; ++ct) {
    const int c0 = ct * 16;
    v2f b = bpos[c0 + l16][half];
    v8f z8 = {};
    v8f d0 = wmma4(a0, b, z8);
    v8f d1 = wmma4(a1, b, z8);
#pragma unroll
    for (int v = 0; v < 8; ++v) {
      dt[wave][(v + 8 * half) * 17 + l16]        = d0[v];
      dt[wave][(v + 8 * half + 16) * 17 + l16]   = d1[v];
    }
    __builtin_amdgcn_wave_barrier();    // wave-private tile: ordering only
#pragma unroll
    for (int j = 0; j < 16; ++j) {
      const int cj = c0 + j;
      float d = dt[wave][lane * 17 + j];
      if (cj != rloc && d < bd[KNN - 1]) {
        float vv = d; int vi = cj;
#pragma unroll
        for (int m = 0; m < KNN; ++m) {          // branchless sorted insert
          const bool  sw = vv < bd[m];
          const float bo = bd[m];
          const int   io = bi[m];
          bd[m] = sw ? vv : bo;
          bi[m] = sw ? vi : io;
          vv    = sw ? bo : vv;
          vi    = sw ? io : vi;
        }
      }
    }
    __builtin_amdgcn_wave_barrier();
  }

  int* op = idx_out + (size_t)(g * NPG + rloc) * KNN;
#pragma unroll
  for (int m = 0; m < KNN; ++m) op[m] = bi[m];
}

// ---------------------------------------------------------------------------
// EdgeConv: one wave per node. A = messages [16 neighbors x 2F] (K padded),
// Linear1 via chained K=4 WMMA (bias in C), leaky, LDS bounce D->A layout,
// Linear2 via 4 chained WMMA, leaky, row-sum over neighbors.
// ---------------------------------------------------------------------------
template <int FIN>
__device__ __forceinline__ float msgval(const float* xi, const float* xj, int c) {
  if (c < FIN)      return xi[c];
  if (c < 2 * FIN)  return xj[c - FIN] - xi[c - FIN];
  return 0.0f;
}

template <int FIN>
__global__ __launch_bounds__(256) void edgeconv_kernel(const float* __restrict__ h,
                                                       const int* __restrict__ idx,
                                                       const float* __restrict__ W1,
                                                       const float* __restrict__ b1,
                                                       const float* __restrict__ W2,
                                                       const float* __restrict__ b2,
                                                       float* __restrict__ out) {
  __shared__ float tl[8][16 * 17];
  const int tid  = threadIdx.x;
  const int wave = tid >> 5;
  const int lane = tid & 31;
  const int half = lane >> 4;
  const int l16  = lane & 15;
  const int node = blockIdx.x * 8 + wave;
  const int g    = node >> 10;
  const int jn   = (g << 10) + idx[(size_t)node * KNN + l16];

  constexpr int KP = (FIN == 3) ? 8 : 32;   // 2F padded to multiple of 4
  constexpr int NT = KP / 4;

  const float* xi = h + (size_t)node * FIN;
  const float* xj = h + (size_t)jn * FIN;

  const float bb1 = b1[l16];
  v8f acc = {bb1, bb1, bb1, bb1, bb1, bb1, bb1, bb1};
#pragma unroll
  for (int t = 0; t < NT; ++t) {
    const int c0 = 4 * t + 2 * half;
    v2f a, b;
    a.x = msgval<FIN>(xi, xj, c0);
    a.y = msgval<FIN>(xi, xj, c0 + 1);
    b.x = (c0 < 2 * FIN) ? W1[c0 * 16 + l16] : 0.0f;
    b.y = (c0 + 1 < 2 * FIN) ? W1[(c0 + 1) * 16 + l16] : 0.0f;
    acc = wmma4(a, b, acc);
  }
#pragma unroll
  for (int v = 0; v < 8; ++v)
    tl[wave][(v + 8 * half) * 17 + l16] = leaky(acc[v]);
  __builtin_amdgcn_wave_barrier();          // wave-private tile: ordering only

  const float bb2 = b2[l16];
  v8f acc2 = {bb2, bb2, bb2, bb2, bb2, bb2, bb2, bb2};
#pragma unroll
  for (int t = 0; t < 4; ++t) {
    const int c0 = 4 * t + 2 * half;
    v2f a, b;
    a.x = tl[wave][l16 * 17 + c0];
    a.y = tl[wave][l16 * 17 + c0 + 1];
    b.x = W2[c0 * 16 + l16];
    b.y = W2[(c0 + 1) * 16 + l16];
    acc2 = wmma4(a, b, acc2);
  }
  float s = 0.0f;
#pragma unroll
  for (int v = 0; v < 8; ++v) s += leaky(acc2[v]);
  s += __shfl_xor(s, 16, 32);               // combine rows 0-7 with 8-15
  if (lane < 16) out[(size_t)node * 16 + lane] = s;
}

// ---------------------------------------------------------------------------
// Post-MLP: cat([x(3), h1(16), h2(16)]) -> 16 -> 16 with leaky. 1 thread/node.
// ---------------------------------------------------------------------------
__global__ __launch_bounds__(256) void post_kernel(const float* __restrict__ x,
                                                   const float* __restrict__ h1,
                                                   const float* __restrict__ h2,
                                                   const float* __restrict__ Wp1,
                                                   const float* __restrict__ bp1,
                                                   const float* __restrict__ Wp2,
                                                   const float* __restrict__ bp2,
                                                   float* __restrict__ hp) {
  const int node = blockIdx.x * 256 + threadIdx.x;
  float cat[35];
#pragma unroll
  for (int c = 0; c < 3; ++c)  cat[c]      = x[(size_t)node * 3 + c];
#pragma unroll
  for (int c = 0; c < 16; ++c) cat[3 + c]  = h1[(size_t)node * 16 + c];
#pragma unroll
  for (int c = 0; c < 16; ++c) cat[19 + c] = h2[(size_t)node * 16 + c];

  float t1[16];
#pragma unroll
  for (int o = 0; o < 16; ++o) {
    float s = bp1[o];
#pragma unroll
    for (int k = 0; k < 35; ++k) s += cat[k] * Wp1[k * 16 + o];
    t1[o] = leaky(s);
  }
#pragma unroll
  for (int o = 0; o < 16; ++o) {
    float s = bp2[o];
#pragma unroll
    for (int k = 0; k < 16; ++k) s += t1[k] * Wp2[k * 16 + o];
    hp[(size_t)node * 16 + o] = leaky(s);
  }
}

// ---------------------------------------------------------------------------
// Pooling: per graph min/max/mean over 1024 nodes of 16 features -> [G,48].
// ---------------------------------------------------------------------------
__global__ __launch_bounds__(256) void pool_kernel(const float* __restrict__ hp,
                                                   float* __restrict__ pooled) {
  __shared__ float smn[256], smx[256], ssm[256];
  const int g   = blockIdx.x;
  const int tid = threadIdx.x;
  const int f   = tid & 15;
  const int seg = tid >> 4;   // 16 segments of 64 nodes
  float mn = 3.0e38f, mx = -3.0e38f, sm = 0.0f;
  for (int i = 0; i < 64; ++i) {
    const int node = g * NPG + seg * 64 + i;
    float v = hp[(size_t)node * 16 + f];
    mn = fminf(mn, v);
    mx = fmaxf(mx, v);
    sm += v;
  }
  smn[tid] = mn; smx[tid] = mx; ssm[tid] = sm;
  __syncthreads();
  if (tid < 16) {
    float amn = 3.0e38f, amx = -3.0e38f, asm_ = 0.0f;
    for (int s = 0; s < 16; ++s) {
      amn = fminf(amn, smn[s * 16 + tid]);
      amx = fmaxf(amx, smx[s * 16 + tid]);
      asm_ += ssm[s * 16 + tid];
    }
    pooled[g * 48 + tid]      = amn;
    pooled[g * 48 + 16 + tid] = amx;
    pooled[g * 48 + 32 + tid] = asm_ * (1.0f / NPG);
  }
}

// ---------------------------------------------------------------------------
// Readout: [G,48] -> leaky(@Wr1+br1) [G,128] -> @Wr2+br2 [G,10].
// ---------------------------------------------------------------------------
__global__ __launch_bounds__(128) void readout_kernel(const float* __restrict__ pooled,
                                                      const float* __restrict__ Wr1,
                                                      const float* __restrict__ br1,
                                                      const float* __restrict__ Wr2,
                                                      const float* __restrict__ br2,
                                                      float* __restrict__ out) {
  __shared__ float hid[128];
  const int g = blockIdx.x;
  const int t = threadIdx.x;
  float s = br1[t];
#pragma unroll
  for (int k = 0; k < 48; ++k) s += pooled[g * 48 + k] * Wr1[k * 128 + t];
  hid[t] = leaky(s);
  __syncthreads();
  if (t < 10) {
    float o = br2[t];
#pragma unroll
    for (int k = 0; k < 128; ++k) o += hid[k] * Wr2[k * 10 + t];
    out[g * 10 + t] = o;
  }
}

// ---------------------------------------------------------------------------
extern "C" void kernel_launch(void* const* d_in, const int* in_sizes, int n_in,
                              void* d_out, int out_size, void* d_ws, size_t ws_size,
                              hipStream_t stream) {
  const float* x   = (const float*)d_in[0];
  const float* W1a = (const float*)d_in[2];
  const float* b1a = (const float*)d_in[3];
  const float* W2a = (const float*)d_in[4];
  const float* b2a = (const float*)d_in[5];
  const float* W1b = (const float*)d_in[6];
  const float* b1b = (const float*)d_in[7];
  const float* W2b = (const float*)d_in[8];
  const float* b2b = (const float*)d_in[9];
  const float* Wp1 = (const float*)d_in[10];
  const float* bp1 = (const float*)d_in[11];
  const float* Wp2 = (const float*)d_in[12];
  const float* bp2 = (const float*)d_in[13];
  const float* Wr1 = (const float*)d_in[14];
  const float* br1 = (const float*)d_in[15];
  const float* Wr2 = (const float*)d_in[16];
  const float* br2 = (const float*)d_in[17];

  char* ws = (char*)d_ws;
  const size_t MB = 1024 * 1024;
  int*   idx1   = (int*)(ws);                  // NN*16*4  = 8 MB
  int*   idx2   = (int*)(ws + 8 * MB);         // 8 MB
  float* h1     = (float*)(ws + 16 * MB);      // 8 MB
  float* h2     = (float*)(ws + 24 * MB);      // 8 MB
  float* hp     = (float*)(ws + 32 * MB);      // 8 MB
  float* pooled = (float*)(ws + 40 * MB);      // 24 KB

  knn_kernel<<<NG * 4, 256, 0, stream>>>(x, 3, idx1);
  edgeconv_kernel<3><<<NN / 8, 256, 0, stream>>>(x, idx1, W1a, b1a, W2a, b2a, h1);
  knn_kernel<<<NG * 4, 256, 0, stream>>>(h1, 16, idx2);
  edgeconv_kernel<16><<<NN / 8, 256, 0, stream>>>(h1, idx2, W1b, b1b, W2b, b2b, h2);
  post_kernel<<<NN / 256, 256, 0, stream>>>(x, h1, h2, Wp1, bp1, Wp2, bp2, hp);
  pool_kernel<<<NG, 256, 0, stream>>>(hp, pooled);
  readout_kernel<<<NG, 128, 0, stream>>>(pooled, Wr1, br1, Wr2, br2, (float*)d_out);
}